// HybridMemory_21921513079409
// MI455X (gfx1250) — compile-verified
//
#include <hip/hip_runtime.h>
#include <hip/hip_bf16.h>
#include <math.h>

// ---------------------------------------------------------------------------
// Problem constants (from reference): B=256, F=256, N=131072, C=2000
//   TEMP = 0.05, EPS = 1e-6
// Whole pipeline reduced to:
//   mem_sum[C,F] = segment_sum(features by labels);  nums[C] = counts
//   simT[B,C]    = (inputs @ mem_sum.T) / TEMP / max(nums,1)      (WMMA f32)
//   loss         = nll + focal + contras + contras_mem  (focal only at target)
// ---------------------------------------------------------------------------

#define CBLK 50          // label-group size: LDS tile = 50*256*4 = 51200 B
#define TEMP_INV 20.0f   // 1 / 0.05
#define EPSV 1e-6f

typedef float v2f __attribute__((ext_vector_type(2)));
typedef float v8f __attribute__((ext_vector_type(8)));

// ---------------- K0: zero mem_sum + nums + d_out --------------------------
__global__ void zero_kernel(float* __restrict__ ws, int n, float* __restrict__ out) {
    int i = blockIdx.x * blockDim.x + threadIdx.x;
    if (i < n) ws[i] = 0.0f;
    if (i == 0) out[0] = 0.0f;
}

// ---------------- K0b: cluster counts --------------------------------------
__global__ void count_kernel(const int* __restrict__ labels, float* __restrict__ nums, int N) {
    int i = blockIdx.x * blockDim.x + threadIdx.x;
    if (i < N) atomicAdd(&nums[labels[i]], 1.0f);
}

// ---------------- K1: segment-sum into per-slice partials ------------------
// grid = (S, G); block = 256 threads (thread == feature column, F must be 256)
__global__ void seg_accum_kernel(const float* __restrict__ features,
                                 const int*  __restrict__ labels,
                                 float* __restrict__ partial,   // [S, C, F]
                                 int N, int F, int C, int S) {
    __shared__ float tile[CBLK * 256];
    __shared__ int   labs[256];
    const int g  = blockIdx.y;
    const int s  = blockIdx.x;
    const int t  = threadIdx.x;
    const int c0 = g * CBLK;
    const int cw = min(CBLK, C - c0);

    for (int i = t; i < CBLK * 256; i += 256) tile[i] = 0.0f;

    const int per = (N + S - 1) / S;
    const int r0  = s * per;
    const int r1  = min(r0 + per, N);

    for (int base = r0; base < r1; base += 256) {
        const int nrow = min(256, r1 - base);
        __syncthreads();
        if (t < nrow) labs[t] = labels[base + t];
        __syncthreads();
        for (int j = 0; j < nrow; ++j) {
            const int lc = labs[j] - c0;
            if ((unsigned)lc < (unsigned)cw) {
                // race-free: rows serialized in-loop, thread t owns column t
                tile[lc * 256 + t] += features[(size_t)(base + j) * F + t];
            }
        }
    }
    __syncthreads();

    float* dst = partial + (size_t)s * C * F + (size_t)c0 * F;
    for (int i = t; i < cw * F; i += 256) dst[i] = tile[i];
}

// ---------------- K1b: reduce partials -> mem_sum --------------------------
__global__ void seg_reduce_kernel(const float* __restrict__ partial,
                                  float* __restrict__ mem_sum,
                                  size_t CF, int S) {
    size_t i = (size_t)blockIdx.x * blockDim.x + threadIdx.x;
    if (i >= CF) return;
    float acc = 0.0f;
    for (int s = 0; s < S; ++s) acc += partial[(size_t)s * CF + i];
    mem_sum[i] = acc;
}

// ---------------- K2: simT = (mem_sum @ inputs.T) * invT / max(nums,1) ----
// One wave per 16(c) x 16(b) tile, V_WMMA_F32_16X16X4_F32, K-loop over F.
__global__ void gemm_sim_kernel(const float* __restrict__ mem_sum,  // [C,F]
                                const float* __restrict__ inputs,   // [B,F]
                                const float* __restrict__ nums,     // [C]
                                float* __restrict__ simT,           // [B,C]
                                int C, int F, int Bn) {
    const int lane  = threadIdx.x & 31;
    const int wave  = threadIdx.x >> 5;
    const int tiles_b = Bn >> 4;
    const int tile  = blockIdx.x * (blockDim.x >> 5) + wave;
    const int cTile = tile / tiles_b;
    const int bTile = tile % tiles_b;
    if (cTile * 16 >= C) return;                 // wave-uniform

    const int lo = lane & 15;
    const int hi = lane >> 4;

    // A: mem_sum rows (M=c, K=f). Lane<16: K={k,k+1}; lane>=16: K={k+2,k+3}.
    const float* Arow = mem_sum + (size_t)(cTile * 16 + lo) * F;
    // B: inputs^T (K=f, N=b): element (K,N) = inputs[N*F + K]; same lane split.
    const float* Brow = inputs  + (size_t)(bTile * 16 + lo) * F;

    v8f acc = {};
    for (int k = 0; k < F; k += 4) {
        const int kk = k + 2 * hi;
        v2f a; a.x = Arow[kk]; a.y = Arow[kk + 1];
        v2f b; b.x = Brow[kk]; b.y = Brow[kk + 1];
        acc = __builtin_amdgcn_wmma_f32_16x16x4_f32(
            /*neg_a=*/false, a, /*neg_b=*/false, b,
            /*c_mod=*/(short)0, acc, /*reuse_a=*/false, /*reuse_b=*/false);
    }

    // D layout: VGPR r holds M = r (lanes 0-15) or r+8 (lanes 16-31), N = lo.
    const int bcol = bTile * 16 + lo;
    #pragma unroll
    for (int r = 0; r < 8; ++r) {
        const int c  = cTile * 16 + r + 8 * hi;
        const float nv = nums[c];
        const float scale = TEMP_INV / (nv > 0.0f ? nv : 1.0f);
        simT[(size_t)bcol * C + c] = acc[r] * scale;
    }
}

// ---------------- K3: per-row losses ---------------------------------------
__device__ __forceinline__ float block_reduce(float v, float* red) {
    const int t = threadIdx.x;
    red[t] = v;
    __syncthreads();
    for (int s = 128; s > 0; s >>= 1) {
        if (t < s) red[t] += red[t + s];
        __syncthreads();
    }
    const float r = red[0];
    __syncthreads();
    return r;
}

__global__ void loss_kernel(const float* __restrict__ inputs,    // [B,F]
                            const float* __restrict__ another,   // [B,F]
                            const float* __restrict__ mem_sum,   // [C,F]
                            const float* __restrict__ nums,      // [C]
                            const float* __restrict__ simT,      // [B,C]
                            const int*  __restrict__ labels,
                            const int*  __restrict__ indexes,
                            float* __restrict__ out,
                            int Bn, int F, int C) {
    __shared__ float red[256];
    const int b   = blockIdx.x;
    const int t   = threadIdx.x;
    const int tgt = labels[indexes[b]];

    // masked sum of exps over the sim row
    float se = 0.0f;
    for (int c = t; c < C; c += 256) {
        const float m = nums[c] > 0.0f ? 1.0f : 0.0f;
        se += m * expf(simT[(size_t)b * C + c]);
    }
    const float sumexp = block_reduce(se, red);

    // F-dim reductions (F == 256 == blockDim)
    float xi = 0.0f, ai = 0.0f, mi = 0.0f;
    if (t < F) {
        xi = inputs[(size_t)b * F + t];
        ai = another[(size_t)b * F + t];
        mi = mem_sum[(size_t)tgt * F + t];
    }
    const float sxx = block_reduce(xi * xi, red);
    const float saa = block_reduce(ai * ai, red);
    const float sxa = block_reduce(xi * ai, red);
    const float smm = block_reduce(mi * mi, red);
    const float sma = block_reduce(mi * ai, red);

    if (t == 0) {
        // mask[tgt] == 1 always (its cluster has >=1 member)
        const float ms_t = expf(simT[(size_t)b * C + tgt]) / (sumexp + EPSV);
        const float l    = logf(ms_t + EPSV);
        const float om   = 1.0f - ms_t;
        const float nll    = -l;
        const float focal  = -(om * om) * (om * om) * l;
        const float contras = -(sxa * rsqrtf(sxx) * rsqrtf(saa));
        // normalize(mem_dyn[tgt]) == normalize(mem_sum[tgt]) (positive scale)
        const float cmem    = -(sma * rsqrtf(smm) * rsqrtf(saa));
        atomicAdd(out, (nll + focal + contras + cmem) * (1.0f / (float)Bn));
    }
}

// ---------------------------------------------------------------------------
extern "C" void kernel_launch(void* const* d_in, const int* in_sizes, int n_in,
                              void* d_out, int out_size, void* d_ws, size_t ws_size,
                              hipStream_t stream) {
    const float* inputs   = (const float*)d_in[0];
    const float* another  = (const float*)d_in[1];
    const float* features = (const float*)d_in[2];
    const int*   labels   = (const int*)d_in[3];
    const int*   indexes  = (const int*)d_in[4];
    // d_in[5] = num_clusters (scalar on device); C fixed by reference = 2000.

    const int N  = in_sizes[3];              // 131072
    const int Bn = in_sizes[4];              // 256
    const int F  = in_sizes[2] / N;          // 256
    const int C  = 2000;
    const size_t CF = (size_t)C * F;

    // workspace layout (floats)
    float* ws       = (float*)d_ws;
    float* mem_sum  = ws;                                    // CF
    float* nums     = ws + CF;                               // C (padded to 64)
    const size_t numsPad = (size_t)((C + 63) & ~63);
    float* simT     = nums + numsPad;                        // Bn*C
    float* partial  = simT + (size_t)Bn * C;                 // S * CF
    const size_t fixedFloats = CF + numsPad + (size_t)Bn * C;

    // choose number of row-slices from available scratch (deterministic)
    size_t availFloats = (ws_size / 4 > fixedFloats) ? (ws_size / 4 - fixedFloats) : 0;
    int S = (int)(availFloats / CF);
    if (S < 1)  S = 1;
    if (S > 64) S = 64;
    const int G = (C + CBLK - 1) / CBLK;     // 40 label groups

    // K0: zero mem_sum + nums (contiguous) + d_out
    {
        const int n = (int)(CF + numsPad);
        zero_kernel<<<(n + 255) / 256, 256, 0, stream>>>(ws, n, (float*)d_out);
    }
    // K0b: cluster counts
    count_kernel<<<(N + 255) / 256, 256, 0, stream>>>(labels, nums, N);
    // K1: partial segment sums
    {
        dim3 grid(S, G);
        seg_accum_kernel<<<grid, 256, 0, stream>>>(features, labels, partial, N, F, C, S);
    }
    // K1b: reduce partials -> mem_sum
    seg_reduce_kernel<<<(int)((CF + 255) / 256), 256, 0, stream>>>(partial, mem_sum, CF, S);
    // K2: WMMA GEMM -> simT
    {
        const int tiles = (C / 16) * (Bn / 16);   // 125 * 16 = 2000 waves
        const int blocks = (tiles + 7) / 8;       // 8 waves / 256-thread block
        gemm_sim_kernel<<<blocks, 256, 0, stream>>>(mem_sum, inputs, nums, simT, C, F, Bn);
    }
    // K3: losses
    loss_kernel<<<Bn, 256, 0, stream>>>(inputs, another, mem_sum, nums, simT,
                                        labels, indexes, (float*)d_out, Bn, F, C);
}